// c_3_75445395521980
// MI455X (gfx1250) — compile-verified
//
#include <hip/hip_runtime.h>
#include <hip/hip_bf16.h>
#include <math.h>

typedef float v2f __attribute__((ext_vector_type(2)));
typedef float v8f __attribute__((ext_vector_type(8)));

// ---------------------------------------------------------------------------
// Kernel 1: conv1 (3->3, 3x3, stride2) + maxpool3 + relu, fused.
// One block per (image, pooled-row). Stages 7 input rows x 3 ch in LDS.
// in:  nodes [320][3][224][224]   out: feat1 [320][3][37][37]
// ---------------------------------------------------------------------------
__global__ __launch_bounds__(256) void k_conv1(const float* __restrict__ nodes,
                                               const float* __restrict__ w,
                                               const float* __restrict__ bias,
                                               float* __restrict__ out) {
    __shared__ __align__(16) float tile[3][7 * 224];   // 4704 floats
    __shared__ float sw[81];
    __shared__ float sb[3];

    const int bid = blockIdx.x;
    const int img = bid / 37;
    const int py  = bid % 37;
    const int tid = threadIdx.x;

    const float* base = nodes + (size_t)img * 150528 + (size_t)(6 * py) * 224;
    // 3 channels x 1568 floats = 3 x 392 float4, all 16B-aligned.
    for (int i = tid; i < 3 * 392; i += 256) {
        const int c = i / 392, j = i % 392;
        const float4* src = (const float4*)(base + (size_t)c * 50176);
        ((float4*)(&tile[c][0]))[j] = src[j];
    }
    if (tid < 81) sw[tid] = w[tid];
    if (tid < 3)  sb[tid] = bias[tid];
    __syncthreads();

    if (tid < 111) {
        const int o = tid / 37, px = tid % 37;
        const float* wo = &sw[o * 27];
        float mx = -1e30f;
#pragma unroll
        for (int ky = 0; ky < 3; ++ky) {
#pragma unroll
            for (int kx = 0; kx < 3; ++kx) {
                float s = sb[o];
                const int r0 = 2 * ky;
                const int c0 = 6 * px + 2 * kx;
#pragma unroll
                for (int ci = 0; ci < 3; ++ci) {
#pragma unroll
                    for (int fy = 0; fy < 3; ++fy) {
                        const float* row = &tile[ci][(r0 + fy) * 224 + c0];
                        const float* wr  = &wo[ci * 9 + fy * 3];
                        s += wr[0] * row[0] + wr[1] * row[1] + wr[2] * row[2];
                    }
                }
                mx = fmaxf(mx, s);
            }
        }
        out[(size_t)img * 4107 + o * 1369 + py * 37 + px] = fmaxf(mx, 0.f);
    }
}

// ---------------------------------------------------------------------------
// Kernel 2: conv2 (3->1, 3x3, stride2) + maxpool3 + relu + linear(36->6).
// One block per image.  in: feat1 [320][3][37][37]  out: gfeat [320][6]
// ---------------------------------------------------------------------------
__global__ __launch_bounds__(64) void k_conv2(const float* __restrict__ in,
                                              const float* __restrict__ w2,
                                              const float* __restrict__ b2,
                                              const float* __restrict__ lw,
                                              const float* __restrict__ lb,
                                              float* __restrict__ feat) {
    __shared__ float t2[3 * 1369];
    __shared__ float sw[27];
    __shared__ float slw[216];
    __shared__ float slb[6];
    __shared__ float sb;
    __shared__ float pooled[36];

    const int img = blockIdx.x, tid = threadIdx.x;
    const float* src = in + (size_t)img * 4107;
    for (int i = tid; i < 4107; i += 64) t2[i] = src[i];
    if (tid < 27) sw[tid] = w2[tid];
    for (int i = tid; i < 216; i += 64) slw[i] = lw[i];
    if (tid < 6)  slb[tid] = lb[tid];
    if (tid == 63) sb = b2[0];
    __syncthreads();

    if (tid < 36) {
        const int py = tid / 6, px = tid % 6;
        float mx = -1e30f;
#pragma unroll
        for (int ky = 0; ky < 3; ++ky) {
#pragma unroll
            for (int kx = 0; kx < 3; ++kx) {
                float s = sb;
                const int r0 = 6 * py + 2 * ky;
                const int c0 = 6 * px + 2 * kx;
#pragma unroll
                for (int ci = 0; ci < 3; ++ci)
#pragma unroll
                    for (int fy = 0; fy < 3; ++fy)
#pragma unroll
                        for (int fx = 0; fx < 3; ++fx)
                            s += sw[ci * 9 + fy * 3 + fx] *
                                 t2[ci * 1369 + (r0 + fy) * 37 + c0 + fx];
                mx = fmaxf(mx, s);
            }
        }
        pooled[tid] = fmaxf(mx, 0.f);
    }
    __syncthreads();
    if (tid < 6) {
        float s = slb[tid];
#pragma unroll
        for (int i = 0; i < 36; ++i) s += pooled[i] * slw[i * 6 + tid];
        feat[img * 6 + tid] = s;
    }
}

// ---------------------------------------------------------------------------
// Kernel 3: graph step + readout MLP. One block, 256 threads = 8 waves.
// fc1/fc2 done with V_WMMA_F32_16X16X4_F32 (M=16 rows fit exactly).
// B-column padding handled with clamped addresses + cndmask (no exec
// predication -> WMMA loop stays branch-free, EXEC stays all-ones).
// ---------------------------------------------------------------------------
__global__ __launch_bounds__(256) void k_graph(
    const float* __restrict__ feat,    // [320][6]
    const float* __restrict__ pos,     // [320][6]
    const float* __restrict__ attmap,  // [80][4][4]
    const float* __restrict__ wfc_w, const float* __restrict__ wfc_b,
    const float* __restrict__ fm_w,  const float* __restrict__ fm_b,
    const float* __restrict__ lm_w,  const float* __restrict__ lm_b,
    const float* __restrict__ fc1_w, const float* __restrict__ fc1_b,
    const float* __restrict__ fc2_w, const float* __restrict__ fc2_b,
    const float* __restrict__ fc3_w, const float* __restrict__ fc3_b,
    float* __restrict__ out) {
    __shared__ float s_feat[320][6];
    __shared__ float s_pos[320][6];
    __shared__ float s_sa[320], s_sb[320];
    __shared__ float s_maw[80][4][4];
    __shared__ float s_pm[320][6];
    __shared__ float s_inp[16][240];
    __shared__ float s_r1[16][128];
    __shared__ float s_r2[16][64];
    __shared__ float s_wfc[24], s_fmw[36], s_fmb[6], s_lmw[36], s_lmb[6];
    __shared__ float s_wfcb;

    const int tid = threadIdx.x;

    for (int i = tid; i < 1920; i += 256) {
        ((float*)s_feat)[i] = feat[i];
        ((float*)s_pos)[i]  = pos[i];
    }
    if (tid < 24) s_wfc[tid] = wfc_w[tid];
    if (tid < 36) { s_fmw[tid] = fm_w[tid]; s_lmw[tid] = lm_w[tid]; }
    if (tid < 6)  { s_fmb[tid] = fm_b[tid]; s_lmb[tid] = lm_b[tid]; }
    if (tid == 255) s_wfcb = wfc_b[0];
    __syncthreads();

    // per-node edge-score halves: sA (source half of wfc), sB (target half)
    for (int n = tid; n < 320; n += 256) {
        float a = 0.f, b = 0.f;
#pragma unroll
        for (int k = 0; k < 6; ++k) {
            a += s_feat[n][k] * s_wfc[k]      + s_pos[n][k] * s_wfc[6 + k];
            b += s_feat[n][k] * s_wfc[12 + k] + s_pos[n][k] * s_wfc[18 + k];
        }
        s_sa[n] = a; s_sb[n] = b;
    }
    // pm = pos @ fm_w + fm_b
    for (int i = tid; i < 1920; i += 256) {
        const int n = i / 6, d = i % 6;
        float s = s_fmb[d];
#pragma unroll
        for (int k = 0; k < 6; ++k) s += s_pos[n][k] * s_fmw[k * 6 + d];
        s_pm[n][d] = s;
    }
    __syncthreads();

    // maw = sigmoid(sA[s]+sB[t]+b) + attmap
    for (int i = tid; i < 1280; i += 256) {
        const int bf = i >> 4, s = (i >> 2) & 3, t = i & 3;
        const float x = s_sa[bf * 4 + s] + s_sb[bf * 4 + t] + s_wfcb;
        s_maw[bf][s][t] = 1.f / (1.f + __expf(-x)) + attmap[i];
    }
    __syncthreads();

    // pos_up = einsum + masked last-frame message; assemble readout input
    for (int i = tid; i < 1920; i += 256) {
        const int d = i % 6, n = (i / 6) & 3, f = (i / 24) % 5, b = i / 120;
        const int bf = b * 5 + f, node = bf * 4 + n;
        float agg = 0.f;
#pragma unroll
        for (int s = 0; s < 4; ++s) agg += s_maw[bf][s][n] * s_pm[bf * 4 + s][d];
        if (f > 0) {
            float lm = s_lmb[d];
            const int pn = (bf - 1) * 4 + n;
#pragma unroll
            for (int k = 0; k < 6; ++k) lm += s_pos[pn][k] * s_lmw[k * 6 + d];
            agg += lm;
        }
        const int col = (f * 4 + n) * 12;
        s_inp[b][col + d]     = s_feat[node][d];
        s_inp[b][col + 6 + d] = agg;
    }
    __syncthreads();

    const int lane = tid & 31, wave = tid >> 5;
    const int m = lane & 15, hi = lane >> 4;

    // fc1: [16,240] @ [240,120] -> 8 waves x 16-col WMMA tiles (pad N to 128)
    {
        const int n = wave * 16 + m;
        const bool nv = (n < 120);
        const int nc = nv ? n : 119;            // clamp: loads always in-bounds
        v8f acc = {};
        for (int k0 = 0; k0 < 240; k0 += 4) {
            const int k = k0 + 2 * hi;          // A 16x4 f32 layout
            v2f a, bb;
            a.x = s_inp[m][k];
            a.y = s_inp[m][k + 1];
            bb.x = fc1_w[k * 120 + nc];          // B 4x16 (K x N)
            bb.y = fc1_w[(k + 1) * 120 + nc];
            if (!nv) { bb.x = 0.f; bb.y = 0.f; } // cndmask, not exec branch
            acc = __builtin_amdgcn_wmma_f32_16x16x4_f32(
                false, a, false, bb, (short)0, acc, false, false);
        }
#pragma unroll
        for (int e = 0; e < 8; ++e) {           // C/D: vgpr=row, lane-half +8
            const int row = e + 8 * hi;
            if (nv) s_r1[row][n] = fmaxf(acc[e] + fc1_b[n], 0.f);
        }
    }
    __syncthreads();

    // fc2: [16,120] @ [120,60] -> waves 0..3 (pad N to 64)
    if (wave < 4) {
        const int n = wave * 16 + m;
        const bool nv = (n < 60);
        const int nc = nv ? n : 59;
        v8f acc = {};
        for (int k0 = 0; k0 < 120; k0 += 4) {
            const int k = k0 + 2 * hi;
            v2f a, bb;
            a.x = s_r1[m][k];
            a.y = s_r1[m][k + 1];
            bb.x = fc2_w[k * 60 + nc];
            bb.y = fc2_w[(k + 1) * 60 + nc];
            if (!nv) { bb.x = 0.f; bb.y = 0.f; }
            acc = __builtin_amdgcn_wmma_f32_16x16x4_f32(
                false, a, false, bb, (short)0, acc, false, false);
        }
#pragma unroll
        for (int e = 0; e < 8; ++e) {
            const int row = e + 8 * hi;
            if (nv) s_r2[row][n] = fmaxf(acc[e] + fc2_b[n], 0.f);
        }
    }
    __syncthreads();

    // fc3: [16,60] @ [60,6] -> 96 outputs, scalar
    if (tid < 96) {
        const int b = tid / 6, j = tid % 6;
        float s = fc3_b[j];
#pragma unroll
        for (int k = 0; k < 60; ++k) s += s_r2[b][k] * fc3_w[k * 6 + j];
        out[tid] = s;
    }
}

// ---------------------------------------------------------------------------
extern "C" void kernel_launch(void* const* d_in, const int* in_sizes, int n_in,
                              void* d_out, int out_size, void* d_ws, size_t ws_size,
                              hipStream_t stream) {
    const float* nodes   = (const float*)d_in[0];
    const float* pos     = (const float*)d_in[1];
    const float* attmap  = (const float*)d_in[2];
    // d_in[3] = depths (unused by the reference)
    const float* conv1_w = (const float*)d_in[4];
    const float* conv1_b = (const float*)d_in[5];
    const float* conv2_w = (const float*)d_in[6];
    const float* conv2_b = (const float*)d_in[7];
    const float* lin_w   = (const float*)d_in[8];
    const float* lin_b   = (const float*)d_in[9];
    const float* wfc_w   = (const float*)d_in[10];
    const float* wfc_b   = (const float*)d_in[11];
    const float* fm_w    = (const float*)d_in[12];
    const float* fm_b    = (const float*)d_in[13];
    const float* lm_w    = (const float*)d_in[14];
    const float* lm_b    = (const float*)d_in[15];
    const float* fc1_w   = (const float*)d_in[16];
    const float* fc1_b   = (const float*)d_in[17];
    const float* fc2_w   = (const float*)d_in[18];
    const float* fc2_b   = (const float*)d_in[19];
    const float* fc3_w   = (const float*)d_in[20];
    const float* fc3_b   = (const float*)d_in[21];

    float* feat1 = (float*)d_ws;                 // 320*3*37*37 floats (~5.3 MB)
    float* gfeat = feat1 + (size_t)320 * 4107;   // 320*6 floats

    k_conv1<<<320 * 37, 256, 0, stream>>>(nodes, conv1_w, conv1_b, feat1);
    k_conv2<<<320, 64, 0, stream>>>(feat1, conv2_w, conv2_b, lin_w, lin_b, gfeat);
    k_graph<<<1, 256, 0, stream>>>(gfeat, pos, attmap,
                                   wfc_w, wfc_b, fm_w, fm_b, lm_w, lm_b,
                                   fc1_w, fc1_b, fc2_w, fc2_b, fc3_w, fc3_b,
                                   (float*)d_out);
}